// ConvKB_2_73065983639796
// MI455X (gfx1250) — compile-verified
//
#include <hip/hip_runtime.h>
#include <math.h>

// ConvKB fused kernel for MI455X (gfx1250, wave32).
// Reference shapes: B=16384, NE=500000, NR=237, K=8, ES=50, TN=4, OC=128.
// conv (K=3 GEMM) via V_WMMA_F32_16X16X4_F32 (K padded to 4), bias through the
// WMMA C operand; relu + fc contraction in VALU (dual-issue max/fmac).
// Gather uses the CDNA5 async global->LDS path (ASYNCcnt) when available.

#define KG      8
#define ES      50
#define TN      4
#define OC      128
#define D_TOT   200     // TN*ES
#define D_PAD   208     // 13 tiles of 16
#define N_FULL  12      // tiles 0..11 fully in-bounds; tile 12 peeled

typedef __attribute__((ext_vector_type(2))) float v2f;
typedef __attribute__((ext_vector_type(8))) float v8f;

// ---------------------------------------------------------------------------
// CDNA5 async global->LDS copy (per-lane LDS scatter address), ASYNCcnt-tracked
// ---------------------------------------------------------------------------
__device__ __forceinline__ void async_load_f32(const float* g, float* l) {
#if __has_builtin(__builtin_amdgcn_global_load_async_to_lds_b32)
  typedef __attribute__((address_space(1))) int GI;   // global
  typedef __attribute__((address_space(3))) int LI;   // LDS
  __builtin_amdgcn_global_load_async_to_lds_b32((GI*)(void*)g, (LI*)(void*)l,
                                                0, 0);
#else
  // VDST = 32-bit LDS address, VADDR = 64-bit global address (GV mode)
  asm volatile("global_load_async_to_lds_b32 %0, %1, off"
               :: "v"((unsigned)(unsigned long long)l), "v"(g) : "memory");
#endif
}

__device__ __forceinline__ void async_wait0() {
#if __has_builtin(__builtin_amdgcn_s_wait_asynccnt)
  __builtin_amdgcn_s_wait_asynccnt(0);
#else
  asm volatile("s_wait_asynccnt 0x0" ::: "memory");
#endif
}

// ---------------------------------------------------------------------------
// Kernel A: per-relation softmax + top-4 (only 237 relations -> one block)
// ---------------------------------------------------------------------------
__global__ __launch_bounds__(256) void relprep_kernel(
    const float* __restrict__ rel_att, int* __restrict__ top_idx,
    float* __restrict__ top_sum, int NR) {
  int r = blockIdx.x * blockDim.x + threadIdx.x;
  if (r >= NR) return;
  float a[8];
  float m = -1e30f;
#pragma unroll
  for (int i = 0; i < 8; ++i) { a[i] = rel_att[r * 8 + i]; m = fmaxf(m, a[i]); }
  float s = 0.f;
#pragma unroll
  for (int i = 0; i < 8; ++i) { a[i] = __expf(a[i] - m); s += a[i]; }
  float inv = 1.f / s;
#pragma unroll
  for (int i = 0; i < 8; ++i) a[i] *= inv;
  unsigned used = 0;
  float ts = 0.f;
  for (int k = 0; k < TN; ++k) {   // descending selection, ties -> lowest idx
    int bi = 0; float bv = -1.f;
#pragma unroll
    for (int i = 0; i < 8; ++i)
      if (!(used & (1u << i)) && a[i] > bv) { bv = a[i]; bi = i; }
    used |= (1u << bi);
    ts += bv;
    top_idx[r * TN + k] = bi;
  }
  top_sum[r] = ts;
}

// ---------------------------------------------------------------------------
// Kernel B: att_loss = 1 - mean(top_sum[rel[b]])
// ---------------------------------------------------------------------------
__global__ __launch_bounds__(256) void attloss_kernel(
    const int* __restrict__ batch, const float* __restrict__ top_sum,
    float* __restrict__ out_loss, int B) {
  __shared__ float red[256];
  float s = 0.f;
  for (int i = threadIdx.x; i < B; i += blockDim.x)
    s += top_sum[batch[3 * i + 1]];
  red[threadIdx.x] = s;
  __syncthreads();
  for (int off = 128; off >= 1; off >>= 1) {
    if ((int)threadIdx.x < off) red[threadIdx.x] += red[threadIdx.x + off];
    __syncthreads();
  }
  if (threadIdx.x == 0) out_loss[0] = 1.0f - red[0] / (float)B;
}

// ---------------------------------------------------------------------------
// wave32 butterfly sum via ds_swizzle (SWAPX16/8/4/2/1)
// ---------------------------------------------------------------------------
__device__ __forceinline__ float wave32_sum(float v) {
  v += __int_as_float(__builtin_amdgcn_ds_swizzle(__float_as_int(v), 0x401F));
  v += __int_as_float(__builtin_amdgcn_ds_swizzle(__float_as_int(v), 0x201F));
  v += __int_as_float(__builtin_amdgcn_ds_swizzle(__float_as_int(v), 0x101F));
  v += __int_as_float(__builtin_amdgcn_ds_swizzle(__float_as_int(v), 0x081F));
  v += __int_as_float(__builtin_amdgcn_ds_swizzle(__float_as_int(v), 0x041F));
  return v;
}

// ---------------------------------------------------------------------------
// Kernel C: fused gather + conv(WMMA) + relu + fc reduction. One block per b.
// 256 threads = 8 waves; wave w owns output channels [16w, 16w+16).
// LDS x layout: sh_k[d][4] = {head_g, rel, tail_g, 0} -> one branchless
// ds_load_b64 per lane supplies the 16x4 f32 A-matrix fragment.
// ---------------------------------------------------------------------------
__global__ __launch_bounds__(256) void convkb_main_kernel(
    const int*   __restrict__ batch,
    const float* __restrict__ entity_emb,
    const float* __restrict__ relation_emb,
    const float* __restrict__ conv_w,
    const float* __restrict__ conv_b,
    const float* __restrict__ fc_w,
    const float* __restrict__ fc_b,
    const int*   __restrict__ top_idx,
    float*       __restrict__ out) {
  __shared__ float sh_k[D_PAD][4];   // [d][k]: k=0 head, 1 rel, 2 tail, 3 zero
  __shared__ float sh_red[8];

  const int b = blockIdx.x;
  const int h = batch[3 * b + 0];
  const int r = batch[3 * b + 1];
  const int t = batch[3 * b + 2];

  // ---- gather into LDS via async global->LDS scatter (k=3 column zeroed) ----
  for (int s = threadIdx.x; s < 4 * D_PAD; s += blockDim.x) {
    const int arr = s / D_PAD;       // 0=head, 1=rel, 2=tail, 3=zero
    const int dd  = s - arr * D_PAD;
    float* lptr = &sh_k[dd][arr];
    if (arr < 3 && dd < D_TOT) {
      const float* g;
      if (arr == 1) {
        g = relation_emb + (long)r * D_TOT + dd;
      } else {
        const int kk = dd / ES;
        const int e  = dd - kk * ES;
        const int gi = top_idx[r * TN + kk];
        g = entity_emb + (long)(arr == 0 ? h : t) * (KG * ES) + gi * ES + e;
      }
      async_load_f32(g, lptr);
    } else {
      *lptr = 0.f;                   // padding (d >= 200 and the K=3 column)
    }
  }
  async_wait0();                     // drain ASYNCcnt before the barrier
  __syncthreads();

  const int wave = threadIdx.x >> 5;
  const int lane = threadIdx.x & 31;
  const int n    = lane & 15;        // tile column -> output channel
  const bool hi  = lane >= 16;
  const int o    = wave * 16 + n;

  // ---- B matrix (conv_w^T, 4x16, K padded): resident for whole block ----
  // f32 4x16 B layout (mirror of 16x4 A): lanes 0-15 -> K=0,1 ; lanes 16-31 -> K=2,3
  v2f bm;
  if (!hi) { bm.x = conv_w[o * 3 + 0]; bm.y = conv_w[o * 3 + 1]; }
  else     { bm.x = conv_w[o * 3 + 2]; bm.y = 0.f; }
  const float bias = conv_b[o];
  v8f cbias;
#pragma unroll
  for (int j = 0; j < 8; ++j) cbias[j] = bias;

  // per-lane base pointers (loop-invariant; tiles become immediate offsets)
  const float* aptr = &sh_k[n][hi ? 2 : 0];                  // A fragment (LDS)
  const float* fptr = fc_w + (long)o * D_TOT + (hi ? 8 : 0); // fc weights (L2)

  float acc = 0.f;
#pragma unroll
  for (int dt = 0; dt < N_FULL; ++dt) {
    const v2f am = *(const v2f*)(aptr + dt * 16 * 4);      // ds_load_b64

    v8f c = __builtin_amdgcn_wmma_f32_16x16x4_f32(
        /*neg_a=*/false, am, /*neg_b=*/false, bm,
        /*c_mod=*/(short)0, cbias, /*reuse_a=*/false, /*reuse_b=*/false);

    // C/D layout: lane holds (d = dt*16 + j [+8 for hi lanes], o = wave*16+n)
    const float4 w0 = *(const float4*)(fptr + dt * 16);
    const float4 w1 = *(const float4*)(fptr + dt * 16 + 4);
    acc = fmaf(c[0] > 0.f ? c[0] : 0.f, w0.x, acc);
    acc = fmaf(c[1] > 0.f ? c[1] : 0.f, w0.y, acc);
    acc = fmaf(c[2] > 0.f ? c[2] : 0.f, w0.z, acc);
    acc = fmaf(c[3] > 0.f ? c[3] : 0.f, w0.w, acc);
    acc = fmaf(c[4] > 0.f ? c[4] : 0.f, w1.x, acc);
    acc = fmaf(c[5] > 0.f ? c[5] : 0.f, w1.y, acc);
    acc = fmaf(c[6] > 0.f ? c[6] : 0.f, w1.z, acc);
    acc = fmaf(c[7] > 0.f ? c[7] : 0.f, w1.w, acc);
  }

  // ---- peeled last tile (d = 192..207): only lo half (192..199) in bounds ----
  {
    const v2f am = *(const v2f*)(aptr + N_FULL * 16 * 4);  // rows 200..207 are 0
    v8f c = __builtin_amdgcn_wmma_f32_16x16x4_f32(
        false, am, false, bm, (short)0, cbias, false, false);
    if (!hi) {
      const float4 w0 = *(const float4*)(fptr + N_FULL * 16);
      const float4 w1 = *(const float4*)(fptr + N_FULL * 16 + 4);
      acc = fmaf(c[0] > 0.f ? c[0] : 0.f, w0.x, acc);
      acc = fmaf(c[1] > 0.f ? c[1] : 0.f, w0.y, acc);
      acc = fmaf(c[2] > 0.f ? c[2] : 0.f, w0.z, acc);
      acc = fmaf(c[3] > 0.f ? c[3] : 0.f, w0.w, acc);
      acc = fmaf(c[4] > 0.f ? c[4] : 0.f, w1.x, acc);
      acc = fmaf(c[5] > 0.f ? c[5] : 0.f, w1.y, acc);
      acc = fmaf(c[6] > 0.f ? c[6] : 0.f, w1.z, acc);
      acc = fmaf(c[7] > 0.f ? c[7] : 0.f, w1.w, acc);
    }
  }

  // ---- reduce: 32 lanes -> 1 (ds_swizzle butterfly), then 8 waves -> 1 ----
  acc = wave32_sum(acc);
  if (lane == 0) sh_red[wave] = acc;
  __syncthreads();
  if (threadIdx.x == 0) {
    float s = 0.f;
#pragma unroll
    for (int w = 0; w < 8; ++w) s += sh_red[w];
    out[b] = s + fc_b[0];
  }
}

// ---------------------------------------------------------------------------
extern "C" void kernel_launch(void* const* d_in, const int* in_sizes, int n_in,
                              void* d_out, int out_size, void* d_ws, size_t ws_size,
                              hipStream_t stream) {
  const int*   batch   = (const int*)d_in[0];
  const float* entity  = (const float*)d_in[1];
  const float* relemb  = (const float*)d_in[2];
  const float* relatt  = (const float*)d_in[3];
  const float* convw   = (const float*)d_in[4];
  const float* convb   = (const float*)d_in[5];
  const float* fcw     = (const float*)d_in[6];
  const float* fcb     = (const float*)d_in[7];
  float* out = (float*)d_out;

  const int B  = in_sizes[0] / 3;
  const int NR = in_sizes[3] / KG;

  int*   ws_top = (int*)d_ws;                      // NR*4 ints
  float* ws_sum = (float*)((char*)d_ws + 4096);    // NR floats

  relprep_kernel<<<(NR + 255) / 256, 256, 0, stream>>>(relatt, ws_top, ws_sum, NR);
  attloss_kernel<<<1, 256, 0, stream>>>(batch, ws_sum, out + B, B);
  convkb_main_kernel<<<B, 256, 0, stream>>>(batch, entity, relemb, convw, convb,
                                            fcw, fcb, ws_top, out);
}